// HypTemporalAttentionLayer_11278584119455
// MI455X (gfx1250) — compile-verified
//
#include <hip/hip_runtime.h>

// MI455X (gfx1250, wave32) implementation.
//
// Roofline: bandwidth-bound. Key trick: expmap0 needs ||M[r,1:]|| where
// M = A.X, A = W.diag(coefU), M[:,0]=0  =>  sum_j M[r,j]^2 = a_r^T G a_r with
// G = X.X^T (16x16 Gram, cols>=1). So ONE pass over X (32MB) via chained
// V_WMMA_F32_16X16X4_F32 (D += Xc.Xc^T) yields both logmap0 row norms (diag G)
// and expmap0 norms. The attention kernel recomputes M chunks on the fly
// (16 FMAs/elem) against L2-resident X (32MB << 192MB L2). HBM traffic
// ~96MB (~4us @ 23.3TB/s) vs ~224MB for the materialized-M version.
// All reductions fixed-order => deterministic. Workspace ~260KB.

typedef __attribute__((ext_vector_type(2))) float v2f;
typedef __attribute__((ext_vector_type(8))) float v8f;

#define EPS_F      1e-7f
#define MIN_NORM_F 1e-15f
#define MAX_NORM_F 1e6f

static constexpr int ROWS = 16;        // t
static constexpr int COLS = 524288;    // n*d (2^19)
// workspace layout (float offsets)
static constexpr int OFF_GPART = 0;      // 256 blocks x 256  = 65536
static constexpr int OFF_AQ    = 65536;  // 16x16 (W_q . diag(coefU))
static constexpr int OFF_AK    = 65792;  // 16x16
static constexpr int OFF_COEFE = 66048;  // 2 tf x 16 rows x 2
static constexpr int OFF_UBQ   = 66112;  // 64 = logmap0(b_q)
static constexpr int OFF_UBK   = 66176;  // 64 = logmap0(b_k)

__device__ __forceinline__ float warp_sum32(float v) {
  v += __shfl_xor(v, 1, 32);
  v += __shfl_xor(v, 2, 32);
  v += __shfl_xor(v, 4, 32);
  v += __shfl_xor(v, 8, 32);
  v += __shfl_xor(v, 16, 32);
  return v;
}
__device__ __forceinline__ float half_sum16(float v) {  // sum within 16-lane halves
  v += __shfl_xor(v, 1, 32);
  v += __shfl_xor(v, 2, 32);
  v += __shfl_xor(v, 4, 32);
  v += __shfl_xor(v, 8, 32);
  return v;
}
__device__ __forceinline__ float arcosh_f(float x) {
  return logf(x + sqrtf(fmaxf(x * x - 1.0f, EPS_F)));
}

// ------ Kernel 1: partial Gram G = X.X^T (cols >= 1) via chained WMMA ------
// 256 blocks x 256 threads; wave handles 64 chunks of 4 columns (256 cols).
// A and B fragments are identical (X tile used as both operands).
__global__ void k_gram(const float* __restrict__ X, float* __restrict__ ws) {
  const int lane = threadIdx.x & 31;
  const int wv = threadIdx.x >> 5;
  const int m = lane & 15;
  const int half = lane >> 4;
  const long rowbase = (long)m * COLS;
  const int c0base = (blockIdx.x * 8 + wv) * 256;

  // warm the next lines of this lane's row
  __builtin_prefetch(X + rowbase + c0base + 2 * half, 0, 3);

  v8f acc = {};
  for (int t = 0; t < 64; ++t) {
    const int c = c0base + t * 4 + 2 * half;
    float2 xv = *reinterpret_cast<const float2*>(X + rowbase + c);  // 8B aligned
    float x0 = (c == 0) ? 0.f : xv.x;   // exclude column 0 (logmap0 y-part)
    v2f f = (v2f){x0, xv.y};            // A[m,2h..2h+1] == B[2h..2h+1,n]
    acc = __builtin_amdgcn_wmma_f32_16x16x4_f32(false, f, false, f, (short)0,
                                                acc, false, false);
  }
  __shared__ float red[8][16][16];
#pragma unroll
  for (int v = 0; v < 8; ++v) red[wv][v + 8 * half][m] = acc[v];
  __syncthreads();
  const int row = threadIdx.x >> 4, col = threadIdx.x & 15;
  float s = 0.f;
  for (int w2 = 0; w2 < 8; ++w2) s += red[w2][row][col];
  ws[OFF_GPART + blockIdx.x * 256 + threadIdx.x] = s;
}

// ------ Kernel 2: all coefficients from G (1 block, 256 threads) -----------
__global__ void k_prep(const float* __restrict__ X, const float* __restrict__ Wq,
                       const float* __restrict__ Wk, const float* __restrict__ bq,
                       const float* __restrict__ bk, const float* __restrict__ cptr,
                       float* __restrict__ ws) {
  __shared__ float G[16][16];
  __shared__ float coefU[16];
  __shared__ float sAQ[256];
  __shared__ float sAK[256];
  __shared__ float tmp[64];
  __shared__ float nrm;
  const int tid = threadIdx.x;
  const float cval = cptr[0];
  const float K = 1.f / cval;
  const float sqrtK = sqrtf(K);

  {  // reduce Gram partials (fixed order)
    float s = 0.f;
    for (int b = 0; b < 256; ++b) s += ws[OFF_GPART + b * 256 + tid];
    G[tid >> 4][tid & 15] = s;
  }
  __syncthreads();
  if (tid < 16) {  // logmap0 row coefficients: diag(G) = ||X[r,1:]||^2
    float yn = fmaxf(sqrtf(G[tid][tid]), MIN_NORM_F);
    float x0 = X[(long)tid * COLS];
    float th = fmaxf(x0 / sqrtK, 1.0f + EPS_F);
    coefU[tid] = sqrtK * arcosh_f(th) / yn;
  }
  __syncthreads();
  {  // a-vectors: A = W . diag(coefU)
    float aq = Wq[tid] * coefU[tid & 15];
    float ak = Wk[tid] * coefU[tid & 15];
    sAQ[tid] = aq; sAK[tid] = ak;
    ws[OFF_AQ + tid] = aq;
    ws[OFF_AK + tid] = ak;
  }
  __syncthreads();
  if (tid < 32) {  // expmap0 coefficients: ||M[r,1:]||^2 = a_r^T G a_r
    const int tf = tid >> 4, r = tid & 15;
    const float* A = (tf ? sAK : sAQ) + r * 16;
    float q = 0.f;
    for (int s = 0; s < 16; ++s) {
      float as = A[s];
      for (int s2 = 0; s2 < 16; ++s2) q += as * A[s2] * G[s][s2];
    }
    float xn = fmaxf(sqrtf(fmaxf(q, 0.f)), MIN_NORM_F);
    float th = xn / sqrtK;
    ws[OFF_COEFE + (tf * 16 + r) * 2 + 0] = sqrtK * coshf(th);       // T[r,0]
    ws[OFF_COEFE + (tf * 16 + r) * 2 + 1] = sqrtK * sinhf(th) / xn;  // scale
  }
  // ub = logmap0(bias) for q and k
  for (int which = 0; which < 2; ++which) {
    const float* b = which ? bk : bq;
    const int off = which ? OFF_UBK : OFF_UBQ;
    if (tid < 64) tmp[tid] = (tid >= 1) ? b[tid] * b[tid] : 0.f;
    __syncthreads();
    if (tid == 0) {
      float s = 0.f;
      for (int j = 1; j < 64; ++j) s += tmp[j];
      nrm = fmaxf(sqrtf(s), MIN_NORM_F);
    }
    __syncthreads();
    if (tid < 64) {
      float th = fmaxf(b[0] / sqrtK, 1.0f + EPS_F);
      float coef = sqrtK * arcosh_f(th) / nrm;
      ws[off + tid] = (tid == 0) ? 0.f : coef * b[tid];
    }
    __syncthreads();
  }
}

// ---- per-64-chunk mobius_add(bias)+proj given raw chunk values (1 wave) ----
__device__ __forceinline__ void mobius_chain(float w0, float w1,
                                             const float* __restrict__ ub,
                                             int lane, float K, float sqrtK,
                                             float* dst) {
  const int e0 = lane * 2, e1 = e0 + 1;
  float x0 = __shfl(w0, 0, 32);
  float m0 = (lane == 0) ? 0.f : w0;           // y-part (dims >= 1)
  float yn2 = warp_sum32(m0 * m0 + w1 * w1);
  float y_norm = fmaxf(sqrtf(yn2), MIN_NORM_F);
  float u0 = ub[e0], u1 = ub[e1];
  float dotyu = warp_sum32(m0 * u0 + w1 * u1);
  float alpha = dotyu / (y_norm * sqrtK);
  float sfac = alpha * (sqrtK - x0) / y_norm;
  float wv0 = (lane == 0) ? (u0 + alpha * y_norm) : (u0 - sfac * w0);
  float wv1 = u1 - sfac * w1;
  float dotywv = warp_sum32(m0 * wv0 + w1 * wv1);          // proj_tan
  float vt0 = (lane == 0) ? (dotywv / fmaxf(x0, EPS_F)) : wv0;
  float vt1 = wv1;
  float vtf = __shfl(vt0, 0, 32);                          // expmap
  float mdot = warp_sum32(vt0 * vt0 + vt1 * vt1) - 2.f * vtf * vtf;
  float mn = fminf(sqrtf(fmaxf(mdot, EPS_F)), MAX_NORM_F);
  float th = fmaxf(mn / sqrtK, MIN_NORM_F);
  float ch = coshf(th), sh = sinhf(th) / th;
  float r0 = ch * w0 + sh * vt0;
  float r1 = ch * w1 + sh * vt1;
  float ry0 = (lane == 0) ? 0.f : r0;                      // proj
  float yn2b = warp_sum32(ry0 * ry0 + r1 * r1);
  float first = sqrtf(fmaxf(K + yn2b, EPS_F));
  dst[e0] = (lane == 0) ? first : r0;
  dst[e1] = r1;
}

// ------ Kernel 3: attention, one block per batch row -----------------------
__global__ void k_attn(const float* __restrict__ X, const float* __restrict__ mask,
                       const float* __restrict__ aptr, const float* __restrict__ cptr,
                       const float* __restrict__ ws, float* __restrict__ out) {
  const int i = blockIdx.x;
  const int tid = threadIdx.x;
  const int lane = tid & 31;
  const int wv = tid >> 5;
  const int r = i >> 9;               // (16,524288)-view row for this batch
  const float cval = cptr[0];
  const float aval = aptr[0];
  const float K = 1.f / cval;
  const float sqrtK = sqrtf(K);

  __shared__ float lq[16][64];
  __shared__ float lk[16][64];
  __shared__ float lv[16][64];        // [..62] valid, [63] = 0 pad
  __shared__ float lamb[16];
  __shared__ float ssum[16];
  __shared__ float sbuf[136];
  __shared__ float subq[64];
  __shared__ float subk[64];
  __shared__ float saq[16];           // a-vector rows for this r
  __shared__ float sak[16];

  if (tid < 64) subq[tid] = ws[OFF_UBQ + tid];
  else if (tid < 128) subk[tid - 64] = ws[OFF_UBK + (tid - 64)];
  else if (tid < 144) saq[tid - 128] = ws[OFF_AQ + r * 16 + (tid - 128)];
  else if (tid < 160) sak[tid - 144] = ws[OFF_AK + r * 16 + (tid - 144)];
  __syncthreads();

  const float cq0 = ws[OFF_COEFE + r * 2 + 0];
  const float cq1 = ws[OFF_COEFE + r * 2 + 1];
  const float ck0 = ws[OFF_COEFE + 32 + r * 2 + 0];
  const float ck1 = ws[OFF_COEFE + 32 + r * 2 + 1];

  const int e0 = lane * 2, e1 = e0 + 1;

  // warm second-rep X span (64 floats x 16 rows)
  {
    const int p2 = i * 16 + wv + 8;
    const long jb2 = (long)(p2 & 8191) * 64;
    __builtin_prefetch(X + (long)(lane & 15) * COLS + jb2 + (lane >> 4) * 32, 0, 3);
  }

  // phase A: recompute M chunk values (M = A.X, L2-resident X), then the
  // mobius_add(bias)+proj chain; wave wv owns times wv and wv+8.
  for (int rep = 0; rep < 2; ++rep) {
    const int tt = wv + rep * 8;
    const int p = i * 16 + tt;
    const long jb = (long)(p & 8191) * 64;   // column base within row r
    float mq0 = 0.f, mq1 = 0.f, mk0 = 0.f, mk1 = 0.f;
#pragma unroll
    for (int s = 0; s < 16; ++s) {
      float2 xv = *reinterpret_cast<const float2*>(X + (long)s * COLS + jb + e0);
      float aqv = saq[s], akv = sak[s];
      mq0 += aqv * xv.x; mq1 += aqv * xv.y;
      mk0 += akv * xv.x; mk1 += akv * xv.y;
    }
    const bool isj0 = (jb == 0) && (lane == 0);  // global column 0: T[r,0]
    float wq0 = isj0 ? cq0 : cq1 * mq0;
    float wq1 = cq1 * mq1;
    float wk0 = isj0 ? ck0 : ck1 * mk0;
    float wk1 = ck1 * mk1;
    mobius_chain(wq0, wq1, subq, lane, K, sqrtK, lq[tt]);
    mobius_chain(wk0, wk1, subk, lane, K, sqrtK, lk[tt]);
  }
  // phase B: v = h2k(big_h), Lorentz factor
  for (int rep = 0; rep < 2; ++rep) {
    const int tt = wv + rep * 8;
    const float* row = X + (long)i * 1024 + tt * 64;
    float h0 = row[0];
    float a0 = (e0 < 63) ? row[1 + e0] / h0 : 0.f;
    float a1 = (e1 < 63) ? row[1 + e1] / h0 : 0.f;
    lv[tt][e0] = a0;
    lv[tt][e1] = a1;
    float sq = warp_sum32(a0 * a0 + a1 * a1);
    if (lane == 0) lamb[tt] = 1.f / sqrtf(1.f - fminf(sq, 0.9f));
  }
  __syncthreads();

  // phase C1: full 16x16 Q.K^T via 16 chained f32 WMMAs (all waves compute
  // redundantly to keep EXEC all-ones; wave 0 publishes to LDS)
  {
    const int nh = lane & 15, half = lane >> 4;
    v8f acc = {};
#pragma unroll
    for (int k = 0; k < 16; ++k) {
      int d0 = 4 * k + 2 * half;
      v2f av = (v2f){lq[nh][d0], lq[nh][d0 + 1]};   // A row m = nh
      v2f bv = (v2f){lk[nh][d0], lk[nh][d0 + 1]};   // B col n = nh (K^T)
      acc = __builtin_amdgcn_wmma_f32_16x16x4_f32(false, av, false, bv,
                                                  (short)0, acc, false, false);
    }
#pragma unroll
    for (int v = 0; v < 8; ++v) {
      int row = v + 8 * half;     // t_m
      float s = 0.f;
      if (nh <= row) {            // triangular pairs only (t_n = nh)
        float md = acc[v] - 2.f * lq[row][0] * lk[nh][0];   // Minkowski dot
        float th = fmaxf(-md * cval, 1.0f + EPS_F);
        float ac = arcosh_f(th);
        float dist = -fminf(K * ac * ac, 50.0f);
        int e = (row * (row + 1)) / 2 + nh;
        s = expf(aval * dist - cval + mask[e]);
        if (wv == 0) sbuf[e] = s;
      }
      float rs = half_sum16(s);   // s_sum[t_m]
      if (wv == 0 && nh == 0) ssum[row] = rs;
    }
  }
  __syncthreads();

  // phase C2: wave wv owns m = wv and m = 15-wv (17 pairs each, balanced)
  for (int g = 0; g < 2; ++g) {
    const int m = g ? (15 - wv) : wv;
    const int eBase = (m * (m + 1)) / 2;
    float z0 = 0.f, z1 = 0.f, wsum = 0.f;
    for (int nn = 0; nn <= m; ++nn) {
      float beta = sbuf[eBase + nn] / ssum[nn];
      float lb = lamb[nn] * beta;
      z0 += lv[nn][e0] * lb;
      z1 += lv[nn][e1] * lb;
      wsum += lb;
    }
    float o0 = z0 / wsum, o1 = z1 / wsum;
    float sq = warp_sum32(o0 * o0 + ((e1 < 63) ? o1 * o1 : 0.f));
    sq = fminf(sq, 0.999f);
    float inv = 1.f / sqrtf(1.f - sq);
    float* orow = out + (long)i * 1024 + m * 64;
    if (lane == 0) orow[0] = inv;          // k2h first coordinate
    orow[1 + e0] = o0 * inv;               // e0 <= 62 always
    if (e1 < 63) orow[1 + e1] = o1 * inv;
  }
}

extern "C" void kernel_launch(void* const* d_in, const int* in_sizes, int n_in,
                              void* d_out, int out_size, void* d_ws, size_t ws_size,
                              hipStream_t stream) {
  const float* bh   = (const float*)d_in[0];
  // d_in[1] = time_index: pairs are exactly [(m,nn) for m in 0..15 for nn<=m],
  // exploited structurally (triangular, grouped by m) — not re-read.
  const float* mask = (const float*)d_in[2];
  const float* wq   = (const float*)d_in[3];
  const float* wk   = (const float*)d_in[4];
  const float* bq   = (const float*)d_in[5];
  const float* bk   = (const float*)d_in[6];
  const float* a    = (const float*)d_in[7];
  const float* c    = (const float*)d_in[8];
  float* ws  = (float*)d_ws;   // ~260 KB used
  float* out = (float*)d_out;

  k_gram<<<256, 256, 0, stream>>>(bh, ws);
  k_prep<<<1, 256, 0, stream>>>(bh, wq, wk, bq, bk, c, ws);
  k_attn<<<8192, 256, 0, stream>>>(bh, mask, a, c, ws, out);
}